// BarycentricCoordinates_83210696393420
// MI455X (gfx1250) — compile-verified
//
#include <hip/hip_runtime.h>
#include <math.h>

typedef __attribute__((ext_vector_type(2))) float v2f;
typedef __attribute__((ext_vector_type(8))) float v8f;

#define NV     4096
#define NB     4
#define K_NBR  32
#define K_LIST 33     // 32 neighbors + the 33rd gives the radius
#define NT     40     // 5 radial * 8 angular template points

// ---------------------------------------------------------------------------
// Kernel 1: async-staged vertices in LDS + WMMA distance strips + top-33.
// One wave32 per 16-row block.
//  - Stage the whole batch vertex array (48KB) into LDS with
//    global_load_async_to_lds_b128, fenced by s_wait_asynccnt.
//  - Loop over 64-column strips: 4 x v_wmma_f32_16x16x4_f32 (K=3 padded)
//    produce a 16x64 tile of dot products; d2 = |vi|^2+|vj|^2-2*dot.
//  - 16 lanes insertion-sort the 33 smallest per row (early reject vs worst).
// ---------------------------------------------------------------------------
__global__ __launch_bounds__(32)
void knn_wmma_kernel(const float* __restrict__ verts,
                     int*   __restrict__ nbrIdx,   // [NB][NV][K_NBR]
                     float* __restrict__ radii)    // [NB][NV]
{
    const int batch = blockIdx.y;
    const int r0    = blockIdx.x * 16;
    const int lane  = threadIdx.x;       // 0..31
    const int m     = lane & 15;
    const int half  = lane >> 4;

    __shared__ alignas(16) float sVerts[NV * 3];   // 48 KB: whole batch
    __shared__ float ldsDist[16][64];              // 4 KB distance strip
    __shared__ float ldsTopD[16][K_LIST];
    __shared__ int   ldsTopI[16][K_LIST];
    __shared__ float ldsSqRow[16];

    const float* vb = verts + (size_t)batch * NV * 3;

    // ---- async stage: 4096*3 floats = 3072 x b128 chunks, striped on lanes
    for (int i = lane; i < (NV * 3) / 4; i += 32) {
        const unsigned ldsOff = (unsigned)(uintptr_t)(&sVerts[i * 4]);
        const float*   gaddr  = vb + (size_t)i * 4;
        asm volatile("global_load_async_to_lds_b128 %0, %1, off"
                     :: "v"(ldsOff), "v"(gaddr) : "memory");
    }
    asm volatile("s_wait_asynccnt 0x0" ::: "memory");
    __syncthreads();

    // Row coordinates -> A fragment (16x4 f32, K=3 padded with 0)
    const int   row = r0 + m;
    const float rx = sVerts[row*3+0];
    const float ry = sVerts[row*3+1];
    const float rz = sVerts[row*3+2];
    if (half == 0) ldsSqRow[m] = rx*rx + ry*ry + rz*rz;

    v2f afrag;
    afrag.x = (half == 0) ? rx : rz;   // VGPR0: K=0 (x) / K=2 (z)
    afrag.y = (half == 0) ? ry : 0.0f; // VGPR1: K=1 (y) / K=3 (pad)

    if (lane < 16) {
        #pragma unroll 1
        for (int j = 0; j < K_LIST; ++j) { ldsTopD[lane][j] = 3.0e38f; ldsTopI[lane][j] = 0; }
    }
    __syncthreads();

    float sqRowReg[8];
    #pragma unroll
    for (int r = 0; r < 8; ++r) sqRowReg[r] = ldsSqRow[r + 8*half];

    for (int c0 = 0; c0 < NV; c0 += 64) {
        // 4 independent WMMAs -> 16x64 distance strip
        #pragma unroll
        for (int u = 0; u < 4; ++u) {
            const int   col = c0 + u*16 + m;
            const float cx = sVerts[col*3+0];
            const float cy = sVerts[col*3+1];
            const float cz = sVerts[col*3+2];
            const float sqc = cx*cx + cy*cy + cz*cz;

            v2f bfrag;                           // B 4x16: N = lane&15
            bfrag.x = (half == 0) ? cx : cz;     // K=0 / K=2
            bfrag.y = (half == 0) ? cy : 0.0f;   // K=1 / K=3

            v8f acc = {};
            acc = __builtin_amdgcn_wmma_f32_16x16x4_f32(
                      false, afrag, false, bfrag, (short)0, acc, false, false);

            // lane holds D[r + 8*half][u*16 + m] in acc[r]
            #pragma unroll
            for (int r = 0; r < 8; ++r) {
                float d2 = sqRowReg[r] + sqc - 2.0f * acc[r];
                d2 = d2 > 0.0f ? d2 : 0.0f;
                ldsDist[r + 8*half][u*16 + m] = sqrtf(d2);
            }
        }
        __syncthreads();

        if (lane < 16) {
            float* dl = ldsTopD[lane];
            int*   il = ldsTopI[lane];
            #pragma unroll 1
            for (int n = 0; n < 64; ++n) {
                const float cand = ldsDist[lane][n];
                if (cand < dl[K_LIST-1]) {           // early reject vs current worst
                    int j = K_LIST - 1;
                    while (j > 0 && dl[j-1] > cand) {
                        dl[j] = dl[j-1]; il[j] = il[j-1]; --j;
                    }
                    dl[j] = cand; il[j] = c0 + n;
                }
            }
        }
        __syncthreads();
    }

    if (lane < 16) {
        const int v = r0 + lane;
        radii[(size_t)batch*NV + v] = ldsTopD[lane][K_LIST-1];
        int* outI = nbrIdx + ((size_t)batch*NV + v) * K_NBR;
        #pragma unroll 1
        for (int k = 0; k < K_NBR; ++k) outI[k] = ldsTopI[lane][k];
    }
}

// ---------------------------------------------------------------------------
// Kernel 2: one wave32 per vertex; lanes <-> 32 neighbors.
// Weighted covariance (butterfly reductions) -> analytic 3x3 eigensolver ->
// LRF + sign disambiguation -> log-map projection -> per-template-point
// 3-closest + barycentric weights.
// ---------------------------------------------------------------------------
__device__ __forceinline__ float waveSum(float v) {
    #pragma unroll
    for (int off = 16; off > 0; off >>= 1)
        v += __shfl_xor(v, off, 32);
    return v;
}

__device__ __forceinline__ void eigvec3(float a00, float a01, float a02,
                                        float a11, float a12, float a22,
                                        float lam,
                                        float& ex, float& ey, float& ez)
{
    const float m00 = a00 - lam, m11 = a11 - lam, m22 = a22 - lam;
    // rows of (A - lam*I): (m00,a01,a02) (a01,m11,a12) (a02,a12,m22)
    const float c0x = a01*a12 - a02*m11, c0y = a02*a01 - m00*a12, c0z = m00*m11 - a01*a01;
    const float c1x = a01*m22 - a02*a12, c1y = a02*a02 - m00*m22, c1z = m00*a12 - a01*a02;
    const float c2x = m11*m22 - a12*a12, c2y = a12*a02 - a01*m22, c2z = a01*a12 - m11*a02;
    const float n0 = c0x*c0x + c0y*c0y + c0z*c0z;
    const float n1 = c1x*c1x + c1y*c1y + c1z*c1z;
    const float n2 = c2x*c2x + c2y*c2y + c2z*c2z;
    float bx = c0x, by = c0y, bz = c0z, bn = n0;
    if (n1 > bn) { bx = c1x; by = c1y; bz = c1z; bn = n1; }
    if (n2 > bn) { bx = c2x; by = c2y; bz = c2z; bn = n2; }
    const float inv = rsqrtf(bn + 1e-30f);
    ex = bx*inv; ey = by*inv; ez = bz*inv;
}

__global__ __launch_bounds__(32)
void bc_kernel(const float* __restrict__ verts,
               const float* __restrict__ tmpl,    // [5][8][2]
               const int*   __restrict__ nbrIdx,
               const float* __restrict__ radii,
               float* __restrict__ out)           // [NB][NV][5][8][3][2]
{
    const int batch = blockIdx.y;
    const int v     = blockIdx.x;
    const int lane  = threadIdx.x;

    __shared__ float sProjX[K_NBR];
    __shared__ float sProjY[K_NBR];
    __shared__ int   sNbr[K_NBR];

    const float* vb = verts + (size_t)batch * NV * 3;
    const float ox = vb[v*3+0], oy = vb[v*3+1], oz = vb[v*3+2];

    const int   nidx = nbrIdx[((size_t)batch*NV + v) * K_NBR + lane];
    const float px = vb[nidx*3+0] - ox;
    const float py = vb[nidx*3+1] - oy;
    const float pz = vb[nidx*3+2] - oz;
    const float dn = sqrtf(px*px + py*py + pz*pz);
    const float rad = radii[(size_t)batch*NV + v];
    const float w = rad - dn;

    // weighted covariance (all lanes end up with the sums)
    const float wsum = waveSum(w);
    float a00 = waveSum(w*px*px);
    float a01 = waveSum(w*px*py);
    float a02 = waveSum(w*px*pz);
    float a11 = waveSum(w*py*py);
    float a12 = waveSum(w*py*pz);
    float a22 = waveSum(w*pz*pz);
    const float inv = 1.0f / (wsum + 1e-8f);
    a00 *= inv; a01 *= inv; a02 *= inv; a11 *= inv; a12 *= inv; a22 *= inv;

    // analytic eigenvalues of symmetric 3x3 (Smith's trigonometric method)
    const float p1 = a01*a01 + a02*a02 + a12*a12;
    const float q  = (a00 + a11 + a22) * (1.0f/3.0f);
    const float p2 = (a00-q)*(a00-q) + (a11-q)*(a11-q) + (a22-q)*(a22-q) + 2.0f*p1;
    const float pp = sqrtf(p2 * (1.0f/6.0f) + 1e-30f);
    const float ip = 1.0f / pp;
    const float b00 = (a00-q)*ip, b11 = (a11-q)*ip, b22 = (a22-q)*ip;
    const float b01 = a01*ip, b02 = a02*ip, b12 = a12*ip;
    float detB = b00*(b11*b22 - b12*b12) - b01*(b01*b22 - b12*b02) + b02*(b01*b12 - b11*b02);
    float rdet = fminf(1.0f, fmaxf(-1.0f, detB * 0.5f));
    const float phi = acosf(rdet) * (1.0f/3.0f);
    const float lamMax = q + 2.0f*pp*cosf(phi);                      // evecs[...,2]
    const float lamMin = q + 2.0f*pp*cosf(phi + 2.0943951023931953f);// evecs[...,0]

    float xx, xy, xz, zx, zy, zz;
    eigvec3(a00,a01,a02,a11,a12,a22, lamMax, xx, xy, xz);
    eigvec3(a00,a01,a02,a11,a12,a22, lamMin, zx, zy, zz);

    // sign disambiguation: s = sum_k sign(dot(neigh_k, axis)); flip if s < 0
    {
        const float dp = px*xx + py*xy + pz*xz;
        const float s  = waveSum((dp > 0.f) ? 1.f : ((dp < 0.f) ? -1.f : 0.f));
        const float f  = (s >= 0.f) ? 1.f : -1.f;
        xx *= f; xy *= f; xz *= f;
    }
    {
        const float dp = px*zx + py*zy + pz*zz;
        const float s  = waveSum((dp > 0.f) ? 1.f : ((dp < 0.f) ? -1.f : 0.f));
        const float f  = (s >= 0.f) ? 1.f : -1.f;
        zx *= f; zy *= f; zz *= f;
    }
    // y = cross(z, x)
    const float yx = zy*xz - zz*xy;
    const float yy = zz*xx - zx*xz;
    const float yz = zx*xy - zy*xx;

    // logarithmic map: plane coords scaled by |neigh| / (|plane| + eps)
    const float ca = px*xx + py*xy + pz*xz;
    const float cb = px*yx + py*yy + pz*yz;
    const float n2p = sqrtf(ca*ca + cb*cb);
    const float sc = dn / (n2p + 1e-8f);
    sProjX[lane] = ca * sc;
    sProjY[lane] = cb * sc;
    sNbr[lane]   = nidx;
    __syncthreads();

    // template points strided over lanes (40 points, wave32)
    for (int t = lane; t < NT; t += 32) {
        const float tx = tmpl[2*t+0];
        const float ty = tmpl[2*t+1];
        float d0 = 3e38f, d1 = 3e38f, d2 = 3e38f;
        int   i0 = 0, i1 = 0, i2 = 0;
        #pragma unroll 1
        for (int k = 0; k < K_NBR; ++k) {
            const float dx = tx - sProjX[k];
            const float dy = ty - sProjY[k];
            const float dd = sqrtf(dx*dx + dy*dy);
            if (dd < d0)      { d2=d1; i2=i1; d1=d0; i1=i0; d0=dd; i0=k; }
            else if (dd < d1) { d2=d1; i2=i1; d1=dd; i1=k; }
            else if (dd < d2) { d2=dd; i2=k; }
        }
        const float p0x = sProjX[i0], p0y = sProjY[i0];
        const float p1x = sProjX[i1], p1y = sProjY[i1];
        const float p2x = sProjX[i2], p2y = sProjY[i2];
        const float v0x = p2x - p0x, v0y = p2y - p0y;
        const float v1x = p1x - p0x, v1y = p1y - p0y;
        const float v2x = tx - p0x,  v2y = ty - p0y;
        const float dot00 = v0x*v0x + v0y*v0y;
        const float dot01 = v0x*v1x + v0y*v1y;
        const float dot02 = v0x*v2x + v0y*v2y;
        const float dot11 = v1x*v1x + v1y*v1y;
        const float dot12 = v1x*v2x + v1y*v2y;
        const float denom = dot00*dot11 - dot01*dot01 + 1e-6f;
        const float w2 = (dot11*dot02 - dot01*dot12) / denom;
        const float w1 = (dot00*dot12 - dot01*dot02) / denom;
        const float w0 = 1.0f - w2 - w1;

        const size_t base = (((size_t)batch*NV + v)*NT + t) * 3 * 2;
        out[base + 0] = (float)sNbr[i0]; out[base + 1] = w2;
        out[base + 2] = (float)sNbr[i1]; out[base + 3] = w1;
        out[base + 4] = (float)sNbr[i2]; out[base + 5] = w0;
    }
}

// ---------------------------------------------------------------------------
extern "C" void kernel_launch(void* const* d_in, const int* in_sizes, int n_in,
                              void* d_out, int out_size, void* d_ws, size_t ws_size,
                              hipStream_t stream) {
    const float* verts = (const float*)d_in[0];   // (4, 4096, 3) f32
    const float* tmpl  = (const float*)d_in[1];   // (5, 8, 2) f32
    float* out = (float*)d_out;                   // (4, 4096, 5, 8, 3, 2) f32

    int*   nbrIdx = (int*)d_ws;                                              // 2 MB
    float* radii  = (float*)((char*)d_ws + (size_t)NB*NV*K_NBR*sizeof(int)); // 64 KB

    dim3 g1(NV/16, NB), b1(32);
    knn_wmma_kernel<<<g1, b1, 0, stream>>>(verts, nbrIdx, radii);

    dim3 g2(NV, NB), b2(32);
    bc_kernel<<<g2, b2, 0, stream>>>(verts, tmpl, nbrIdx, radii, out);
}